// BIODecoder_29265907155094
// MI455X (gfx1250) — compile-verified
//
#include <hip/hip_runtime.h>

// CRF NLL for B=256, T=2048, K=41 on gfx1250 (MI455X, wave32).
//
// Z-scan trick: logsumexp_i(alpha_i + M_ij) = m + log( sum_i exp(alpha_i-m) * expT_ij )
// with expT = exp(transitions) precomputed once. Each step is then a
// 16x44 * 44x48 fp32 GEMM per 16-row batch tile -> V_WMMA_F32_16X16X4_F32.
// Emissions for step t+1 are staged into LDS with GLOBAL_LOAD_ASYNC_TO_LDS_B32
// (ASYNCcnt) while step t computes, hiding HBM latency in the serial scan.

typedef float v2f __attribute__((ext_vector_type(2)));
typedef float v8f __attribute__((ext_vector_type(8)));

#define K_TAGS 41
#define KPAD   44   // K rounded up to multiple of 4 (WMMA f32 k-step = 4)
#define NPAD   48   // K rounded up to 3 tiles of 16
#define TILE_B 16
#define FWD_THREADS 96  // 3 waves: one 16-wide N-tile each

#ifndef __has_builtin
#define __has_builtin(x) 0
#endif

#if __has_builtin(__builtin_amdgcn_global_load_async_to_lds_b32)
#define HAVE_ASYNC_LDS 1
typedef __attribute__((address_space(1))) int gint_t;   // global int
typedef __attribute__((address_space(3))) int lint_t;   // LDS int
__device__ __forceinline__ void async_copy_b32(const float* gp, float* lp) {
    __builtin_amdgcn_global_load_async_to_lds_b32((gint_t*)gp, (lint_t*)lp, 0, 0);
}
__device__ __forceinline__ void async_wait0() {
#if __has_builtin(__builtin_amdgcn_s_wait_asynccnt)
    __builtin_amdgcn_s_wait_asynccnt(0);
#else
    asm volatile("s_wait_asynccnt 0" ::: "memory");
#endif
}
#else
#define HAVE_ASYNC_LDS 0
__device__ __forceinline__ void async_copy_b32(const float* gp, float* lp) { *lp = *gp; }
__device__ __forceinline__ void async_wait0() {}
#endif

// ---------------------------------------------------------------------------
// Gold-path score: posterior[b] = start[t0] + sum(trans + em along tagged path) + end[tlast]
// ---------------------------------------------------------------------------
__global__ void crf_path_kernel(const float* __restrict__ emissions,
                                const unsigned char* __restrict__ mask,
                                const long long* __restrict__ tags,
                                const float* __restrict__ start_tr,
                                const float* __restrict__ trans,
                                const float* __restrict__ end_tr,
                                float* __restrict__ post_out,
                                int B, int T) {
    int b = blockIdx.x * blockDim.x + threadIdx.x;
    if (b >= B) return;
    const size_t bT = (size_t)b * T;
    long long prev = tags[bT];
    float s = start_tr[prev] + emissions[bT * K_TAGS + (size_t)prev];
    for (int t = 1; t < T; ++t) {
        if (mask[bT + t]) {
            long long cur = tags[bT + t];
            s += trans[(size_t)prev * K_TAGS + cur] +
                 emissions[(bT + t) * K_TAGS + (size_t)cur];
            prev = cur;
        }
    }
    s += end_tr[prev];
    post_out[b] = s;
}

// ---------------------------------------------------------------------------
// Forward partition scan. One workgroup = 16 batch rows, 3 waves (one N-tile each).
// ---------------------------------------------------------------------------
__global__ __launch_bounds__(FWD_THREADS)
void crf_forward_kernel(const float* __restrict__ emissions,
                        const unsigned char* __restrict__ mask,
                        const float* __restrict__ start_tr,
                        const float* __restrict__ trans,
                        const float* __restrict__ end_tr,
                        float* __restrict__ z_out,
                        int B, int T) {
    __shared__ float expT[KPAD][NPAD];      // exp(transitions), zero-padded
    __shared__ float alpha[TILE_B][NPAD];   // current alpha, one row per batch element
    __shared__ float expA[TILE_B][KPAD];    // exp(alpha - rowmax), zero-padded k
    __shared__ __align__(16) float emLDS[2][TILE_B][KPAD];  // double-buffered emissions
    __shared__ float mrow[TILE_B];          // per-row max
    __shared__ unsigned char mcur[TILE_B];  // per-row mask at current t

    const int tid  = threadIdx.x;       // 0..95
    const int wave = tid >> 5;          // 0..2 -> N-tile
    const int lane = tid & 31;
    const int ln   = lane & 15;         // N index within tile / M index for A-frag
    const int lh   = lane >> 4;         // half-wave selector
    const int n0   = wave * 16;
    const int bbase = blockIdx.x * TILE_B;

    // expT[i][j] = exp(trans[i][j]) inside 41x41, else 0 so pads contribute nothing.
    for (int idx = tid; idx < KPAD * NPAD; idx += FWD_THREADS) {
        int i = idx / NPAD, j = idx % NPAD;
        expT[i][j] = (i < K_TAGS && j < K_TAGS) ? __expf(trans[i * K_TAGS + j]) : 0.f;
    }
    // alpha0 = start + em[t=0]
    for (int idx = tid; idx < TILE_B * NPAD; idx += FWD_THREADS) {
        int r = idx / NPAD, j = idx % NPAD;
        float v = -1e30f;
        if (j < K_TAGS)
            v = start_tr[j] + emissions[((size_t)(bbase + r) * T) * K_TAGS + j];
        alpha[r][j] = v;
    }
    // Stage emissions for t=1 into buffer 1 (async, B32 granularity: K=41 is odd so
    // row starts are only 4B-aligned in global memory).
    for (int idx = tid; idx < TILE_B * K_TAGS; idx += FWD_THREADS) {
        int r = idx / K_TAGS, k = idx - r * K_TAGS;
        async_copy_b32(&emissions[((size_t)(bbase + r) * T + 1) * K_TAGS + k],
                       &emLDS[1][r][k]);
    }
    async_wait0();
    __syncthreads();

    for (int t = 1; t < T; ++t) {
        const int cur = t & 1;
        const int nxt = cur ^ 1;

        // Kick off async staging of next step's emissions; overlaps this whole step.
        if (t + 1 < T) {
            for (int idx = tid; idx < TILE_B * K_TAGS; idx += FWD_THREADS) {
                int r = idx / K_TAGS, k = idx - r * K_TAGS;
                async_copy_b32(&emissions[((size_t)(bbase + r) * T + t + 1) * K_TAGS + k],
                               &emLDS[nxt][r][k]);
            }
        }

        // Per-row max + mask byte (threads 0..15).
        if (tid < TILE_B) {
            float m = alpha[tid][0];
            #pragma unroll 8
            for (int j = 1; j < K_TAGS; ++j) m = fmaxf(m, alpha[tid][j]);
            mrow[tid] = m;
            mcur[tid] = mask[(size_t)(bbase + tid) * T + t];
        }
        __syncthreads();

        // expA[r][k] = exp(alpha[r][k] - mrow[r]); pad k -> 0
        for (int idx = tid; idx < TILE_B * KPAD; idx += FWD_THREADS) {
            int r = idx / KPAD, k = idx % KPAD;
            expA[r][k] = (k < K_TAGS) ? __expf(alpha[r][k] - mrow[r]) : 0.f;
        }
        __syncthreads();

        // GEMM: S = expA(16x44) * expT(44x48), this wave computes cols [n0, n0+16).
        // A frag (16x4 f32): lane ln=M, VGPR v holds K = k0 + 2*lh + v.
        // B frag (4x16 f32): lane ln=N, VGPR v holds K = k0 + 2*lh + v.
        v8f c = {};
        #pragma unroll
        for (int kk = 0; kk < KPAD / 4; ++kk) {
            const int k0 = kk * 4;
            v2f a, b;
            a[0] = expA[ln][k0 + 2 * lh + 0];
            a[1] = expA[ln][k0 + 2 * lh + 1];
            b[0] = expT[k0 + 2 * lh + 0][n0 + ln];
            b[1] = expT[k0 + 2 * lh + 1][n0 + ln];
            c = __builtin_amdgcn_wmma_f32_16x16x4_f32(
                    /*neg_a=*/false, a, /*neg_b=*/false, b,
                    /*c_mod=*/(short)0, c, /*reuse_a=*/false, /*reuse_b=*/false);
        }
        __syncthreads();  // everyone done reading expA/alpha before we overwrite alpha

        // D frag 16x16 f32: VGPR r holds row M = r + 8*lh, col N = ln.
        // Emissions for this step come from LDS (staged last iteration).
        const int j = n0 + ln;
        if (j < K_TAGS) {
            #pragma unroll
            for (int r = 0; r < 8; ++r) {
                int row = r + 8 * lh;
                if (mcur[row]) {
                    float e = emLDS[cur][row][j];
                    alpha[row][j] = mrow[row] + __logf(c[r]) + e;
                }
            }
        }
        async_wait0();    // our t+1 staging complete before anyone reads it next iter
        __syncthreads();
    }

    // z[b] = logsumexp_j(alpha[b] + end[j])
    if (tid < TILE_B) {
        float m = -1e30f;
        for (int j = 0; j < K_TAGS; ++j) m = fmaxf(m, alpha[tid][j] + end_tr[j]);
        float s = 0.f;
        for (int j = 0; j < K_TAGS; ++j) s += __expf(alpha[tid][j] + end_tr[j] - m);
        z_out[bbase + tid] = m + __logf(s);
    }
}

// ---------------------------------------------------------------------------
// Final mean reduction: out = mean(posterior - z)
// ---------------------------------------------------------------------------
__global__ void crf_reduce_kernel(const float* __restrict__ post,
                                  const float* __restrict__ z,
                                  float* __restrict__ out, int B) {
    __shared__ float sdata[256];
    int tid = threadIdx.x;
    sdata[tid] = (tid < B) ? (post[tid] - z[tid]) : 0.f;
    __syncthreads();
    for (int s = 128; s > 0; s >>= 1) {
        if (tid < s) sdata[tid] += sdata[tid + s];
        __syncthreads();
    }
    if (tid == 0) out[0] = sdata[0] / (float)B;
}

extern "C" void kernel_launch(void* const* d_in, const int* in_sizes, int n_in,
                              void* d_out, int out_size, void* d_ws, size_t ws_size,
                              hipStream_t stream) {
    const float*         emissions = (const float*)d_in[0];
    const unsigned char* mask      = (const unsigned char*)d_in[1];   // jax bool = 1 byte
    const long long*     tags      = (const long long*)d_in[2];       // int64
    const float*         start_tr  = (const float*)d_in[3];
    const float*         trans     = (const float*)d_in[4];
    const float*         end_tr    = (const float*)d_in[5];

    const int B = 256;
    const int T = in_sizes[1] / B;   // mask is [B,T]

    float* post = (float*)d_ws;      // [B]
    float* z    = post + B;          // [B]

    crf_path_kernel<<<(B + 63) / 64, 64, 0, stream>>>(
        emissions, mask, tags, start_tr, trans, end_tr, post, B, T);
    crf_forward_kernel<<<B / TILE_B, FWD_THREADS, 0, stream>>>(
        emissions, mask, start_tr, trans, end_tr, z, B, T);
    crf_reduce_kernel<<<1, 256, 0, stream>>>(post, z, (float*)d_out, B);
}